// ModelNew_5531917877383
// MI455X (gfx1250) — compile-verified
//
#include <hip/hip_runtime.h>

// ---- CDNA5 TDM descriptor vector types (clang-23 6-arg builtin form) ----
typedef __attribute__((ext_vector_type(4))) unsigned int tdm_v4u;
typedef __attribute__((ext_vector_type(8))) int          tdm_v8i;
typedef __attribute__((ext_vector_type(4))) int          tdm_v4i;
typedef __attribute__((ext_vector_type(4))) float        f4;

#define ROW_LEN        32768              // elements per row (scan axis)
#define N_ROWS         2048
#define TPB            1024               // 32 waves (wave32)
#define CHUNK          32                 // floats per thread = one row / 1024
#define GROUP_DW       36                 // 32 data dwords + 4 pad dwords (TDM pad)
#define BUF_DW         (TPB * GROUP_DW)   // 36864 dwords = 144 KB per row buffer
#define SCRATCH_DW     64                 // scan scratch (keeps buffers 256B aligned)
#define N_BLOCKS       512
#define ROWS_PER_BLOCK (N_ROWS / N_BLOCKS)

// Issue a TDM DMA of one full row (32768 f32) from global into LDS at ldsByteAddr,
// inserting 4 pad dwords after every 32 dwords so per-thread chunks land at LDS
// dword stride 36 (bank stride 36 mod 64 -> only 2-way conflicts on ds_read_b128).
__device__ __forceinline__ void tdm_load_row(const float* __restrict__ rowPtr,
                                             unsigned ldsByteAddr) {
  unsigned long long ga = (unsigned long long)(const void*)rowPtr;

  tdm_v4u g0;
  g0.x = 1u;                                            // count=1, is_restore=0, gather=0
  g0.y = ldsByteAddr;                                   // lds_addr (bytes, wave-relative)
  g0.z = (unsigned)ga;                                  // global_addr[31:0]
  g0.w = (((unsigned)(ga >> 32)) & 0x01FFFFFFu)         // global_addr[56:32]
       | (2u << 30);                                    // type = 2 ("image")

  tdm_v8i g1;
  g1[0] = (int)((2u << 16)      // data_size = 2 -> 4 bytes
              | (1u << 20)      // pad_enable
              | (4u << 22)      // pad_interval = 4 -> every 32 dwords
              | (3u << 25));    // pad_amount   = 3 -> 4 dwords
  g1[1] = (int)0x80000000u;     // tensor_dim0[15:0]=0x8000 (32768) in bits 63:48
  g1[2] = (int)0x00010000u;     // tensor_dim0 hi = 0 ; tensor_dim1 lo16 = 1
  g1[3] = (int)0x80000000u;     // tensor_dim1 hi16 = 0 ; tile_dim0 = 0x8000 (32768)
  g1[4] = 1;                    // tile_dim1 = 1, tile_dim2 = 0
  g1[5] = ROW_LEN;              // tensor_dim0_stride low 32
  g1[6] = 0;                    // stride0 hi16 = 0 ; tensor_dim1_stride lo16 = 0
  g1[7] = 0;

  tdm_v4i g2 = {0, 0, 0, 0};    // 2D tensor: groups 2/3 unused -> 2-SGPR-group form
  tdm_v4i g3 = {0, 0, 0, 0};
  tdm_v8i g4 = {0, 0, 0, 0, 0, 0, 0, 0};   // extra group in 6-arg form: unused

  __builtin_amdgcn_tensor_load_to_lds(g0, g1, g2, g3, g4, 0);
}

__global__ void __launch_bounds__(TPB, 1)
cumprod_rows(const float* __restrict__ x, float* __restrict__ out) {
  extern __shared__ float lds[];          // [0..63] scan scratch, then 2 row buffers

  const int t    = threadIdx.x;
  const int lane = t & 31;
  const int wave = t >> 5;

  int row = blockIdx.x;

  // Prime the pipeline: DMA first row into buffer 0 (one TDM issue, wave 0 only).
  if (wave == 0) tdm_load_row(x + (size_t)row * ROW_LEN, SCRATCH_DW * 4u);

  int buf = 0;
  for (int it = 0; it < ROWS_PER_BLOCK; ++it) {
    // Wait for the DMA filling the current buffer, then publish to all waves.
    if (wave == 0) __builtin_amdgcn_s_wait_tensorcnt(0);
    __syncthreads();   // also guarantees everyone finished reading the other buffer

    // Overlap: kick off DMA of the next row into the other buffer.
    if (wave == 0 && it + 1 < ROWS_PER_BLOCK) {
      tdm_load_row(x + (size_t)(row + N_BLOCKS) * ROW_LEN,
                   (unsigned)((SCRATCH_DW + (buf ^ 1) * BUF_DW) * 4u));
    }

    // ---- load this thread's 32-float chunk from LDS (padded layout) ----
    const unsigned base = SCRATCH_DW + (unsigned)buf * BUF_DW + (unsigned)t * GROUP_DW;
    const f4* src = (const f4*)&lds[base];
    float l[CHUNK];
#pragma unroll
    for (int j = 0; j < CHUNK / 4; ++j) {
      f4 q = src[j];
      l[4 * j + 0] = q.x; l[4 * j + 1] = q.y;
      l[4 * j + 2] = q.z; l[4 * j + 3] = q.w;
    }

    // ---- log2 and per-thread sum (v_log_f32 is native log2) ----
    float s = 0.0f;
#pragma unroll
    for (int k = 0; k < CHUNK; ++k) {
      l[k] = __builtin_amdgcn_logf(l[k]);
      s += l[k];
    }

    // ---- wave32 inclusive shuffle scan of per-thread sums ----
    float incl = s;
#pragma unroll
    for (int d = 1; d < 32; d <<= 1) {
      float n = __shfl_up(incl, (unsigned)d, 32);
      if (lane >= d) incl += n;
    }
    if (lane == 31) lds[wave] = incl;     // wave totals -> scratch
    __syncthreads();

    // ---- wave 0 scans the 32 wave totals into exclusive wave offsets ----
    if (wave == 0) {
      float ws = lds[lane];
      float winc = ws;
#pragma unroll
      for (int d = 1; d < 32; d <<= 1) {
        float n = __shfl_up(winc, (unsigned)d, 32);
        if (lane >= d) winc += n;
      }
      lds[lane] = winc - ws;              // exclusive offset per wave
    }
    __syncthreads();

    // ---- exclusive prefix for this thread, then accumulate + exp2 + store ----
    float acc = lds[wave] + (incl - s);

    f4* dst = (f4*)(out + (size_t)row * ROW_LEN + (size_t)t * CHUNK);
#pragma unroll
    for (int j = 0; j < CHUNK / 4; ++j) {
      f4 q;
      acc += l[4 * j + 0]; q.x = __builtin_amdgcn_exp2f(acc);
      acc += l[4 * j + 1]; q.y = __builtin_amdgcn_exp2f(acc);
      acc += l[4 * j + 2]; q.z = __builtin_amdgcn_exp2f(acc);
      acc += l[4 * j + 3]; q.w = __builtin_amdgcn_exp2f(acc);
      __builtin_nontemporal_store(q, dst + j);   // stream-once output: th:NT
    }

    buf ^= 1;
    row += N_BLOCKS;
  }
}

extern "C" void kernel_launch(void* const* d_in, const int* in_sizes, int n_in,
                              void* d_out, int out_size, void* d_ws, size_t ws_size,
                              hipStream_t stream) {
  (void)in_sizes; (void)n_in; (void)out_size; (void)d_ws; (void)ws_size;
  const float* x = (const float*)d_in[0];
  float* out = (float*)d_out;

  const size_t shmem = (size_t)(SCRATCH_DW + 2 * BUF_DW) * 4;   // 295,168 B <= 320 KB
  (void)hipFuncSetAttribute((const void*)cumprod_rows,
                            hipFuncAttributeMaxDynamicSharedMemorySize, (int)shmem);
  cumprod_rows<<<N_BLOCKS, TPB, shmem, stream>>>(x, out);
}